// HiRALayer_3977139716930
// MI455X (gfx1250) — compile-verified
//
#include <hip/hip_runtime.h>
#include <hip/hip_bf16.h>

#if defined(__HIP_DEVICE_COMPILE__)
#if !__has_builtin(__builtin_amdgcn_wmma_f32_16x16x4_f32)
#error "device pass: missing __builtin_amdgcn_wmma_f32_16x16x4_f32"
#endif
#endif

typedef __attribute__((ext_vector_type(2))) float v2f;
typedef __attribute__((ext_vector_type(8))) float v8f;

// ---------------------------------------------------------------------------
// Stage 1: Wa[o,i] = W[o,i] * (1 + sum_r A[r,i] * B[o,r])
// ---------------------------------------------------------------------------
__global__ __launch_bounds__(256) void hira_adapt_weight(
    const float* __restrict__ W,
    const float* __restrict__ A,
    const float* __restrict__ Bl,
    float* __restrict__ Wa,
    int IN, int R) {
  __shared__ float bs[32];
  const int o = blockIdx.y;
  const int i = blockIdx.x * 256 + threadIdx.x;
  if (threadIdx.x < (unsigned)R) bs[threadIdx.x] = Bl[o * R + threadIdx.x];
  __syncthreads();
  float acc = 1.0f;
#pragma unroll 16
  for (int r = 0; r < R; ++r) acc += A[(size_t)r * IN + i] * bs[r];
  const size_t idx = (size_t)o * IN + i;
  Wa[idx] = W[idx] * acc;
}

// ---------------------------------------------------------------------------
// Stage 2: Out[m,n] = sum_k X[m,k] * Wa[n,k] + bias[n]
// 128x128 block tile, K-chunk 32, 8 wave32s (4Mx2N), 2x4 16x16 acc tiles per
// wave, fp32 WMMA 16x16x4. Double-buffered LDS with register staging: next
// chunk's global_load_b128s issue before the current chunk's 64 WMMAs, so
// s_wait_loadcnt lands after the compute instead of on the critical path.
// One barrier per chunk.
// ---------------------------------------------------------------------------
#define TM 128
#define TN 128
#define TK 32
#define LDSS 36   // padded row stride in floats: 8B-aligned, bank-conflict-free

__global__ __launch_bounds__(256) void hira_gemm_f32_wmma(
    const float* __restrict__ X,    // [M, K]
    const float* __restrict__ Wa,   // [N, K]
    const float* __restrict__ bias, // [N]
    float* __restrict__ Out,        // [M, N]
    int M, int N, int K) {
  __shared__ float lA[2][TM * LDSS];
  __shared__ float lB[2][TN * LDSS];

  const int tid  = threadIdx.x;
  const int lane = tid & 31;
  const int wave = tid >> 5;
  const int wm   = wave >> 1;  // 0..3 : wave row (32 M-rows each)
  const int wn   = wave & 1;   // 0..1 : wave col (64 N-cols each)

  const int m0 = blockIdx.y * TM;
  const int n0 = blockIdx.x * TN;

  const int half = lane >> 4;  // K-pair selector for fp32 fragments
  const int l16  = lane & 15;  // row-within-tile for A/B fragments

  v8f acc[2][4];
#pragma unroll
  for (int ti = 0; ti < 2; ++ti)
#pragma unroll
    for (int tj = 0; tj < 4; ++tj) acc[ti][tj] = (v8f)0.0f;

  // cooperative tile load mapping: 256 threads, float4 each, 32 rows/pass
  const int ldRow = tid >> 3;        // 0..31
  const int ldCol = (tid & 7) * 4;   // 0,4,...,28

  float4 rx[4], rw[4];  // register staging for the in-flight chunk

  auto gload = [&](int k0) {
#pragma unroll
    for (int p = 0; p < 4; ++p) {
      const int row = p * 32 + ldRow;
      rx[p] = *(const float4*)(X + (size_t)(m0 + row) * K + (k0 + ldCol));
      rw[p] = *(const float4*)(Wa + (size_t)(n0 + row) * K + (k0 + ldCol));
    }
  };
  auto lstore = [&](int buf) {
#pragma unroll
    for (int p = 0; p < 4; ++p) {
      const int row = p * 32 + ldRow;
      *(float4*)(&lA[buf][row * LDSS + ldCol]) = rx[p];
      *(float4*)(&lB[buf][row * LDSS + ldCol]) = rw[p];
    }
  };

  gload(0);
  lstore(0);
  __syncthreads();

  const int NC = K / TK;
  for (int c = 0; c < NC; ++c) {
    const int buf = c & 1;
    if (c + 1 < NC) gload((c + 1) * TK);  // prefetch next chunk (overlaps WMMAs)

#if defined(__HIP_DEVICE_COMPILE__)
#pragma unroll
    for (int kk = 0; kk < TK; kk += 4) {
      v2f afrag[2];
      v2f bfrag[4];
#pragma unroll
      for (int ti = 0; ti < 2; ++ti)
        afrag[ti] = *(const v2f*)(&lA[buf][(wm * 32 + ti * 16 + l16) * LDSS +
                                           (kk + 2 * half)]);
#pragma unroll
      for (int tj = 0; tj < 4; ++tj)
        bfrag[tj] = *(const v2f*)(&lB[buf][(wn * 64 + tj * 16 + l16) * LDSS +
                                           (kk + 2 * half)]);
#pragma unroll
      for (int ti = 0; ti < 2; ++ti)
#pragma unroll
        for (int tj = 0; tj < 4; ++tj)
          acc[ti][tj] = __builtin_amdgcn_wmma_f32_16x16x4_f32(
              /*neg_a=*/false, afrag[ti],
              /*neg_b=*/false, bfrag[tj],
              /*c_mod=*/(short)0, acc[ti][tj],
              /*reuse_a=*/false, /*reuse_b=*/false);
    }
#endif

    if (c + 1 < NC) lstore(buf ^ 1);  // fill the other buffer; readers of it
                                      // finished at the previous barrier
    __syncthreads();
  }

  // Epilogue: C/D layout is VGPR v -> row (v + 8*half), lane%16 -> col.
#pragma unroll
  for (int ti = 0; ti < 2; ++ti) {
#pragma unroll
    for (int tj = 0; tj < 4; ++tj) {
      const int n  = n0 + wn * 64 + tj * 16 + l16;
      const float bv = bias[n];
      const int mbase = m0 + wm * 32 + ti * 16 + 8 * half;
#pragma unroll
      for (int v = 0; v < 8; ++v) {
        Out[(size_t)(mbase + v) * N + n] = acc[ti][tj][v] + bv;
      }
    }
  }
}

extern "C" void kernel_launch(void* const* d_in, const int* in_sizes, int n_in,
                              void* d_out, int out_size, void* d_ws,
                              size_t ws_size, hipStream_t stream) {
  const float* x    = (const float*)d_in[0];  // [B,S,IN] -> [M,K]
  const float* W    = (const float*)d_in[1];  // [OUT,IN]
  const float* bias = (const float*)d_in[2];  // [OUT]
  const float* lA   = (const float*)d_in[3];  // [R,IN]
  const float* lB   = (const float*)d_in[4];  // [OUT,R]
  float* out = (float*)d_out;

  const int OUT = in_sizes[2];
  const int IN  = in_sizes[1] / OUT;
  const int M   = in_sizes[0] / IN;
  const int R   = in_sizes[3] / IN;

  float* Wa = (float*)d_ws;  // OUT*IN fp32 scratch (64 MiB for this shape)

  dim3 g1(IN / 256, OUT);
  hira_adapt_weight<<<g1, 256, 0, stream>>>(W, lA, lB, Wa, IN, R);

  dim3 g2(OUT / TN, M / TM);
  hira_gemm_f32_wmma<<<g2, 256, 0, stream>>>(x, Wa, bias, out, M, OUT, IN);
}